// FCOS_11141145166405
// MI455X (gfx1250) — compile-verified
//
#include <hip/hip_runtime.h>

// ---------------------------------------------------------------------------
// MI455X (gfx1250) implementation of Revision_PRED-style box refinement.
//  K1: pairwise IoU, one 16x16 tile per wave32; V_WMMA_F32_16X16X4_F32
//      produces the (area1_i + area2_j) tile exactly (rank-2 broadcast sum
//      as a K=4 f32 matmul), VALU does the overlap min/max math.
//  K2 x8: per-column max/argmax over the 134MB ious matrix (L2-resident on
//      MI455X's 192MB L2) + the single-element-per-column suppression write.
//  K3: per-column finalize (refined boxes / scores / classes).
//  K4: per-row missing_mask reduction (wave-per-row, coalesced).
// ---------------------------------------------------------------------------

typedef float v2f __attribute__((ext_vector_type(2)));
typedef float v8f __attribute__((ext_vector_type(8)));

#define N1 8192
#define N2 4096
#define NUM_ITERS 8
#define COL_TILES (N2 / 16)   // 256
#define ROW_TILES (N1 / 16)   // 512

// ---------------- Kernel 1: IoU matrix via WMMA area-sum tiles -------------
__global__ __launch_bounds__(256) void k_iou(const float* __restrict__ b1,
                                             const float* __restrict__ b2,
                                             float* __restrict__ ious) {
    const int wid  = (blockIdx.x * 256 + threadIdx.x) >> 5;   // global wave id
    const int lane = threadIdx.x & 31;
    const int tr = wid / COL_TILES;
    const int tc = wid % COL_TILES;
    const int rb = tr * 16, cb = tc * 16;
    const int half = lane >> 4;        // 0: lanes 0-15, 1: lanes 16-31
    const int l    = lane & 15;

    const float4* B1 = (const float4*)b1;
    const float4* B2 = (const float4*)b2;

    // Areas for this tile's rows/cols (lanes 16-31 load same rows; cached).
    float4 bx = B1[rb + l];
    float4 by = B2[cb + l];
    float area1 = (bx.z - bx.x) * (bx.w - bx.y);
    float area2 = (by.z - by.x) * (by.w - by.y);

    // A (16x4, f32): lane<16 holds {K0=area1, K1=1}; lane>=16 holds {K2=0,K3=0}
    // B (4x16, f32): lane<16 holds {K0=1, K1=area2}; lane>=16 holds {K2=0,K3=0}
    // => D[i][j] = area1[i]*1 + 1*area2[j], exact in f32 RNE.
    float m = half ? 0.0f : 1.0f;
    v2f A; A.x = m * area1; A.y = m;
    v2f B; B.x = m;         B.y = m * area2;

    v8f C = {};
    C = __builtin_amdgcn_wmma_f32_16x16x4_f32(false, A, false, B, (short)0, C,
                                              false, false);

    // C/D layout: VGPR r, lanes 0-15 -> (row rb+r, col cb+lane);
    //             lanes 16-31 -> (row rb+r+8, col cb+lane-16).
    const int col = cb + l;
    float4 q = B2[col];
#pragma unroll
    for (int r = 0; r < 8; ++r) {
        int row = rb + r + 8 * half;
        float4 p = B1[row];
        float ix = fmaxf(fminf(p.z, q.z) - fmaxf(p.x, q.x), 0.0f);
        float iy = fmaxf(fminf(p.w, q.w) - fmaxf(p.y, q.y), 0.0f);
        float ov = ix * iy;
        float un = fmaxf(C[r] - ov, 1e-6f);
        ious[(long)row * N2 + col] = ov / un;
    }
}

// --------------- Per-column segmented argmax (first-occurrence) ------------
__device__ __forceinline__ void col_argmax_seg(const float* __restrict__ ious,
                                               int col, int seg,
                                               float& best, int& bidx) {
    const int SEG = N1 / 8;            // 1024 rows per segment thread
    const int r0  = seg * SEG;
    best = -1.0f;
    bidx = 0;
    for (int i = 0; i < SEG; ++i) {    // strict > keeps lowest row index
        float v = ious[(long)(r0 + i) * N2 + col];
        if (v > best) { best = v; bidx = r0 + i; }
    }
}

// --------------- Kernel 2: one suppression iteration (run 8x) --------------
__global__ __launch_bounds__(256) void k_colstep(float* __restrict__ ious,
                                                 const float* __restrict__ s1,
                                                 const float* __restrict__ s2) {
    __shared__ float sv[256];
    __shared__ int   si[256];
    const int t   = threadIdx.x;
    const int c   = t & 31;            // 32 columns per block
    const int seg = t >> 5;            // 8 row-segments per column
    const int col = blockIdx.x * 32 + c;

    float best; int bidx;
    col_argmax_seg(ious, col, seg, best, bidx);
    sv[t] = best; si[t] = bidx;
    __syncthreads();

    if (seg == 0) {
        for (int k = 1; k < 8; ++k) {  // in segment order: first-occurrence
            float v = sv[k * 32 + c];
            if (v > best) { best = v; bidx = si[k * 32 + c]; }
        }
        // lower = refine_gt & !(scores1[argmax] >= scores2); write 0.5.
        if (best > 0.5f && !(s1[bidx] >= s2[col]))
            ious[(long)bidx * N2 + col] = 0.5f;
    }
}

// --------------- Kernel 3: per-column outputs ------------------------------
__global__ __launch_bounds__(256) void k_colfinal(
        const float* __restrict__ ious,
        const float* __restrict__ bb1, const float* __restrict__ bb2,
        const float* __restrict__ s1,  const float* __restrict__ s2,
        const int* __restrict__ c1,    const int* __restrict__ c2,
        float* __restrict__ out_rb, float* __restrict__ out_s2,
        float* __restrict__ out_c2) {
    __shared__ float sv[256];
    __shared__ int   si[256];
    const int t   = threadIdx.x;
    const int c   = t & 31;
    const int seg = t >> 5;
    const int col = blockIdx.x * 32 + c;

    float best; int bidx;
    col_argmax_seg(ious, col, seg, best, bidx);
    sv[t] = best; si[t] = bidx;
    __syncthreads();

    if (seg == 0) {
        for (int k = 1; k < 8; ++k) {
            float v = sv[k * 32 + c];
            if (v > best) { best = v; bidx = si[k * 32 + c]; }
        }
        const bool refine = best > 0.5f;
        const float* src = refine ? (bb1 + 4 * bidx) : (bb2 + 4 * col);
#pragma unroll
        for (int k = 0; k < 4; ++k) {
            float b = src[k];
            out_rb[col * 4 + k] = (b + fabsf(b)) * 0.5f;   // clamp negatives
        }
        out_s2[col] = refine ? s1[bidx] : s2[col];
        out_c2[col] = (float)(refine ? c1[bidx] : c2[col]);
    }
}

// --------------- Kernel 4: missing_mask = all(ious[row,:] < 0.5) -----------
__global__ __launch_bounds__(256) void k_rows(const float* __restrict__ ious,
                                              float* __restrict__ mm) {
    const int lane = threadIdx.x & 31;
    const int w    = threadIdx.x >> 5;
    const int row  = blockIdx.x * 8 + w;       // wave per row, coalesced scan
    const float* p = ious + (long)row * N2;
    int bad = 0;
    for (int j = lane; j < N2; j += 32) bad |= (p[j] >= 0.5f) ? 1 : 0;
#pragma unroll
    for (int off = 16; off; off >>= 1) bad |= __shfl_down(bad, off, 32);
    if (lane == 0) mm[row] = bad ? 0.0f : 1.0f;
}

// ---------------------------------------------------------------------------
extern "C" void kernel_launch(void* const* d_in, const int* in_sizes, int n_in,
                              void* d_out, int out_size, void* d_ws, size_t ws_size,
                              hipStream_t stream) {
    const float* bboxes1 = (const float*)d_in[0];   // [8192,4]
    const float* scores1 = (const float*)d_in[1];   // [8192]
    const int*   classes1= (const int*)  d_in[2];   // [8192]
    const float* bboxes2 = (const float*)d_in[3];   // [4096,4]
    const float* scores2 = (const float*)d_in[4];   // [4096]
    const int*   classes2= (const int*)  d_in[5];   // [4096]

    float* out = (float*)d_out;
    float* out_rb  = out;                    // 16384
    float* out_s2  = out + 16384;            // 4096
    float* out_c2  = out + 20480;            // 4096
    float* out_mm  = out + 24576;            // 8192
    float* ious    = out + 32768;            // 8192*4096, computed in place

    // K1: 131072 tiles (512x256), 8 waves/block -> 16384 blocks.
    k_iou<<<(ROW_TILES * COL_TILES) / 8, 256, 0, stream>>>(bboxes1, bboxes2, ious);

    // K2: 8 bounded suppression iterations (L2-resident passes).
    for (int it = 0; it < NUM_ITERS; ++it)
        k_colstep<<<N2 / 32, 256, 0, stream>>>(ious, scores1, scores2);

    // K3/K4: finalize.
    k_colfinal<<<N2 / 32, 256, 0, stream>>>(ious, bboxes1, bboxes2,
                                            scores1, scores2, classes1, classes2,
                                            out_rb, out_s2, out_c2);
    k_rows<<<N1 / 8, 256, 0, stream>>>(ious, out_mm);
}